// ViTMoE_11802570130366
// MI455X (gfx1250) — compile-verified
//
#include <hip/hip_runtime.h>
#include <hip/hip_bf16.h>

// ---------------------------------------------------------------------------
// ViT-MoE forward, gfx1250 (MI455X), wave32.
//
// Exact algebraic reduction: the reference output is z[:,0] @ head_w.T +
// head_b with z = LN(moe), and moe[:,0] depends only on
// t[:,0] = cls_token + pos_embed[0,0,:] -- batch independent. The image, the
// patch-embed GEMM and all 196 patch tokens are dead code. We compute one cls
// vector through its top-2 experts (VALU matvecs, one 128-thread block), then
// a 16x128 x 128x1000 head GEMM with V_WMMA_F32_16X16X4_F32 (fully unrolled,
// 4 independent accumulator chains), A staged into LDS via
// GLOBAL_LOAD_ASYNC_TO_LDS_B128 + s_wait_asynccnt. Row 0 of D is broadcast to
// all 128 identical batch rows.
// ---------------------------------------------------------------------------

#define E 128
#define HID 512
#define NEXP 8
#define NCLS 1000
#define EPS 1e-5f

typedef __attribute__((ext_vector_type(2))) float v2f;
typedef __attribute__((ext_vector_type(8))) float v8f;

__device__ __forceinline__ float block_sum128(float v, float* buf, int tid) {
    buf[tid] = v;
    __syncthreads();
#pragma unroll
    for (int s = 64; s > 0; s >>= 1) {
        if (tid < s) buf[tid] += buf[tid + s];
        __syncthreads();
    }
    float r = buf[0];
    __syncthreads();
    return r;
}

// One 128-thread block (4 wave32s on one WGP). Computes z[0..127] into ws.
__global__ void __launch_bounds__(128)
cls_moe_kernel(const float* __restrict__ cls_token, const float* __restrict__ pos_embed,
               const float* __restrict__ router_w,  const float* __restrict__ router_b,
               const float* __restrict__ ln1_g, const float* __restrict__ ln1_b,
               const float* __restrict__ wv,    const float* __restrict__ bv,
               const float* __restrict__ wo,    const float* __restrict__ bo,
               const float* __restrict__ ln2_g, const float* __restrict__ ln2_b,
               const float* __restrict__ w1,    const float* __restrict__ b1,
               const float* __restrict__ w2,    const float* __restrict__ b2,
               const float* __restrict__ norm_g, const float* __restrict__ norm_b,
               float* __restrict__ zout) {
    const int tid = threadIdx.x;
    __shared__ float sT[E];      // t0 (cls token embedding), never overwritten
    __shared__ float sXn[E];     // shared LN1 stats applied
    __shared__ float sBuf[E];    // stage buffer (y / a / hn)
    __shared__ float sMh[HID];   // gelu(ffn1)
    __shared__ float sRed[E];    // reduction scratch
    __shared__ int   sIdx[2];    // top-2 expert indices

    // t[:,0,:] = cls_token + pos_embed[0,0,:]  (batch independent)
    float t0 = cls_token[tid] + pos_embed[tid];
    sT[tid] = t0;
    __syncthreads();

    // Router logits (softmax is monotone -> top-k of logits == top-k of scores)
    if (tid < NEXP) {
        float s = router_b[tid];
        const float* rw = router_w + tid * E;
#pragma unroll 8
        for (int e = 0; e < E; ++e) s += sT[e] * rw[e];
        sRed[tid] = s;
    }
    __syncthreads();
    if (tid == 0) {
        // top-2, ties -> lowest index (matches jax.lax.top_k)
        int i0 = 0; float v0 = sRed[0];
        for (int x = 1; x < NEXP; ++x)
            if (sRed[x] > v0) { v0 = sRed[x]; i0 = x; }
        int i1 = -1; float v1 = -3.402823466e38f;
        for (int x = 0; x < NEXP; ++x)
            if (x != i0 && sRed[x] > v1) { v1 = sRed[x]; i1 = x; }
        sIdx[0] = i0; sIdx[1] = i1;
    }
    __syncthreads();

    // Shared LN1 statistics over t0
    float m1 = block_sum128(t0, sRed, tid) * (1.0f / E);
    float d1 = t0 - m1;
    float var1 = block_sum128(d1 * d1, sRed, tid) * (1.0f / E);
    float xn = d1 * __frsqrt_rn(var1 + EPS);
    sXn[tid] = xn;
    __syncthreads();

    float acc = 0.0f;
#pragma unroll 1
    for (int k = 0; k < 2; ++k) {
        const int x = sIdx[k];
        // y = xn * ln1_g + ln1_b
        float y = sXn[tid] * ln1_g[x * E + tid] + ln1_b[x * E + tid];
        sBuf[tid] = y;
        __syncthreads();

        // a = y @ wv[x].T + bv   (wv[x][f][e], f = tid)
        const float* wvx = wv + (size_t)x * E * E + (size_t)tid * E;
        __builtin_prefetch(wvx, 0, 1);  // global_prefetch_b8
        float a = bv[x * E + tid];
#pragma unroll 8
        for (int e = 0; e < E; ++e) a += sBuf[e] * wvx[e];
        __syncthreads();
        sBuf[tid] = a;
        __syncthreads();

        // a2 = a @ wo[x].T + bo
        const float* wox = wo + (size_t)x * E * E + (size_t)tid * E;
        float a2 = bo[x * E + tid];
#pragma unroll 8
        for (int f = 0; f < E; ++f) a2 += sBuf[f] * wox[f];

        // residual + LN2
        float h = sT[tid] + a2;
        float m2 = block_sum128(h, sRed, tid) * (1.0f / E);
        float d2 = h - m2;
        float var2 = block_sum128(d2 * d2, sRed, tid) * (1.0f / E);
        float hn = d2 * __frsqrt_rn(var2 + EPS) * ln2_g[x * E + tid] + ln2_b[x * E + tid];
        __syncthreads();
        sBuf[tid] = hn;
        __syncthreads();

        // mh = gelu(hn @ w1[x] + b1), exact erf gelu. w1[x][e][j], j = tid + 128*r
        const float* w1x = w1 + (size_t)x * E * HID;
#pragma unroll
        for (int r = 0; r < 4; ++r) {
            const int j = tid + E * r;
            float s = b1[x * HID + j];
#pragma unroll 8
            for (int e = 0; e < E; ++e) s += sBuf[e] * w1x[e * HID + j];
            sMh[j] = 0.5f * s * (1.0f + erff(s * 0.70710678118654752f));
        }
        __syncthreads();

        // out = h + mh @ w2[x] + b2   (w2[x][j][e], e = tid)
        const float* w2x = w2 + (size_t)x * HID * E + tid;
        float o = h + b2[x * E + tid];
#pragma unroll 8
        for (int j = 0; j < HID; ++j) o += sMh[j] * w2x[(size_t)j * E];
        acc += o;
        __syncthreads();
    }

    // unweighted mean of the 2 experts, final LN, write z
    float moe = acc * 0.5f;
    float m3 = block_sum128(moe, sRed, tid) * (1.0f / E);
    float d3 = moe - m3;
    float var3 = block_sum128(d3 * d3, sRed, tid) * (1.0f / E);
    zout[tid] = d3 * __frsqrt_rn(var3 + EPS) * norm_g[tid] + norm_b[tid];
}

// Head GEMM via V_WMMA_F32_16X16X4_F32: A = z replicated into 16 rows,
// B = head_w.T tile (128 x 16). One wave per 16-column tile, K fully
// unrolled into 4 independent accumulation chains. z is staged into LDS with
// one GLOBAL_LOAD_ASYNC_TO_LDS_B128 per lane (32 x 16B = 512B), fenced by
// s_wait_asynccnt 0. D row 0 (VGPR 0, lanes 0-15) holds the 16 logits of the
// tile; each lane broadcasts its logit to all 128 (identical) batch rows.
__global__ void __launch_bounds__(32)
head_wmma_kernel(const float* __restrict__ z, const float* __restrict__ head_w,
                 const float* __restrict__ head_b, float* __restrict__ out) {
    __shared__ float sZ[E];

    const int tile = blockIdx.x;        // 0..62
    const int lane = threadIdx.x;       // 0..31
    const bool hi  = lane >= 16;
    const int  n   = hi ? lane - 16 : lane;   // column-in-tile owned by this lane (B/D frag)
    const int  col = tile * 16 + n;
    const int  colc = col < NCLS ? col : (NCLS - 1);  // clamp: OOB columns compute garbage,
                                                      // are never stored (per-column independence)
    const float* __restrict__ bcol = head_w + (size_t)colc * E;

    // Async-stage z (512 B) into LDS: lane L copies bytes [16L, 16L+16).
    // Generic shared-pointer low 32 bits == LDS byte address (aperture
    // mapping truncates to addr[31:0]).
    {
        unsigned lds_addr = (unsigned)(unsigned long long)(&sZ[0]) + (unsigned)(lane * 16);
        unsigned goff     = (unsigned)(lane * 16);
        asm volatile("global_load_async_to_lds_b128 %0, %1, %2"
                     :
                     : "v"(lds_addr), "v"(goff), "s"(z)
                     : "memory");
        asm volatile("s_wait_asynccnt 0x0" ::: "memory");
        // single wave is the only consumer -> no workgroup barrier needed
    }

    const int koff = hi ? 2 : 0;  // A 16x4 f32 frag: lanes 0-15 hold K=+0,+1; lanes 16-31 K=+2,+3
    v8f c0 = {}, c1 = {}, c2 = {}, c3 = {};

#pragma unroll
    for (int k0 = 0; k0 < E; k0 += 16) {
        // four independent D->C accumulation chains, K-steps k0, k0+4, k0+8, k0+12
        {
            v2f a; a.x = sZ[k0 + koff];     a.y = sZ[k0 + koff + 1];
            v2f b; b.x = bcol[k0 + koff];   b.y = bcol[k0 + koff + 1];
            c0 = __builtin_amdgcn_wmma_f32_16x16x4_f32(false, a, false, b, (short)0, c0, false, false);
        }
        {
            v2f a; a.x = sZ[k0 + 4 + koff];   a.y = sZ[k0 + 4 + koff + 1];
            v2f b; b.x = bcol[k0 + 4 + koff]; b.y = bcol[k0 + 4 + koff + 1];
            c1 = __builtin_amdgcn_wmma_f32_16x16x4_f32(false, a, false, b, (short)0, c1, false, false);
        }
        {
            v2f a; a.x = sZ[k0 + 8 + koff];   a.y = sZ[k0 + 8 + koff + 1];
            v2f b; b.x = bcol[k0 + 8 + koff]; b.y = bcol[k0 + 8 + koff + 1];
            c2 = __builtin_amdgcn_wmma_f32_16x16x4_f32(false, a, false, b, (short)0, c2, false, false);
        }
        {
            v2f a; a.x = sZ[k0 + 12 + koff];   a.y = sZ[k0 + 12 + koff + 1];
            v2f b; b.x = bcol[k0 + 12 + koff]; b.y = bcol[k0 + 12 + koff + 1];
            c3 = __builtin_amdgcn_wmma_f32_16x16x4_f32(false, a, false, b, (short)0, c3, false, false);
        }
    }
    v8f c = (c0 + c1) + (c2 + c3);

    if (lane < 16 && col < NCLS) {
        const float val = c[0] + head_b[col];  // D[M=0][N=lane]
#pragma unroll 4
        for (int bi = 0; bi < 128; ++bi) out[bi * NCLS + col] = val;
    }
}

extern "C" void kernel_launch(void* const* d_in, const int* in_sizes, int n_in,
                              void* d_out, int out_size, void* d_ws, size_t ws_size,
                              hipStream_t stream) {
    (void)in_sizes; (void)n_in; (void)out_size; (void)ws_size;
    // setup_inputs() order:
    // 0 x, 1 conv_w, 2 conv_b, 3 cls_token, 4 pos_embed, 5 router_w, 6 router_b,
    // 7 ln1_g, 8 ln1_b, 9 wv, 10 bv, 11 wo, 12 bo, 13 ln2_g, 14 ln2_b,
    // 15 w1, 16 b1, 17 w2, 18 b2, 19 norm_g, 20 norm_b, 21 head_w, 22 head_b
    const float* cls_token = (const float*)d_in[3];
    const float* pos_embed = (const float*)d_in[4];
    const float* router_w  = (const float*)d_in[5];
    const float* router_b  = (const float*)d_in[6];
    const float* ln1_g     = (const float*)d_in[7];
    const float* ln1_b     = (const float*)d_in[8];
    const float* wv        = (const float*)d_in[9];
    const float* bv        = (const float*)d_in[10];
    const float* wo        = (const float*)d_in[11];
    const float* bo        = (const float*)d_in[12];
    const float* ln2_g     = (const float*)d_in[13];
    const float* ln2_b     = (const float*)d_in[14];
    const float* w1        = (const float*)d_in[15];
    const float* b1        = (const float*)d_in[16];
    const float* w2        = (const float*)d_in[17];
    const float* b2        = (const float*)d_in[18];
    const float* norm_g    = (const float*)d_in[19];
    const float* norm_b    = (const float*)d_in[20];
    const float* head_w    = (const float*)d_in[21];
    const float* head_b    = (const float*)d_in[22];

    float* zbuf = (float*)d_ws;  // 128 floats of scratch

    cls_moe_kernel<<<dim3(1), dim3(128), 0, stream>>>(
        cls_token, pos_embed, router_w, router_b, ln1_g, ln1_b,
        wv, bv, wo, bo, ln2_g, ln2_b, w1, b1, w2, b2, norm_g, norm_b, zbuf);

    head_wmma_kernel<<<dim3((NCLS + 15) / 16), dim3(32), 0, stream>>>(
        zbuf, head_w, head_b, (float*)d_out);
}